// FeatureDenoising_73529840107641
// MI455X (gfx1250) — compile-verified
//
#include <hip/hip_runtime.h>
#include <hip/hip_bf16.h>

typedef float v2f __attribute__((ext_vector_type(2)));
typedef float v8f __attribute__((ext_vector_type(8)));

#define BB 16
#define CC 256
#define LL 2048
#define KT 16          // K tile staged in LDS per buffer (pointwise)
#define KT2 16         // K tile staged in LDS per buffer (attn_apply)
#define FS_PAD 20      // padded LDS row (floats): 80B rows, 16B aligned, conflict-free
#define BN_EPS 1e-5f

static __device__ __forceinline__ v8f wmma_f32(v2f a, v2f b, v8f c) {
    // D = A(16x4) * B(4x16) + C ; fp32 WMMA
    return __builtin_amdgcn_wmma_f32_16x16x4_f32(
        /*neg_a=*/false, a, /*neg_b=*/false, b,
        /*c_mod=*/(short)0, c, /*reuse_a=*/false, /*reuse_b=*/false);
}

static __device__ __forceinline__ unsigned lds_addr_of(const void* p) {
    // generic (shared) pointer -> 32-bit workgroup-relative LDS byte address
    return (unsigned)(size_t)(__attribute__((address_space(3))) const void*)p;
}

static __device__ __forceinline__ void async_copy_b128(unsigned lds_dst,
                                                       const void* gsrc) {
    asm volatile("global_load_async_to_lds_b128 %0, %1, off"
                 :: "v"(lds_dst), "v"((unsigned long long)(size_t)gsrc)
                 : "memory");
}

// ---------------------------------------------------------------------------
// Pointwise conv (1x1): Out[b,o,l] = sum_c W[o,c] * X[b,c,l] + bias[o]
// GEMM: M=CC (o), K=CC (c), N=BB*LL. Block 256 thr = 8 waves (2Mx4N),
// wave tile 64x64 -> block tile 128x256. X tile (KTx256) staged in LDS via
// double-buffered async copies; W read directly (L2-resident, 256 KB).
// ---------------------------------------------------------------------------
__global__ __launch_bounds__(256) void pointwise_gemm(
    const float* __restrict__ X, const float* __restrict__ W,
    const float* __restrict__ bias, float* __restrict__ Out) {
    __shared__ float xs[2][KT * 256];   // 2 x 16 KB

    const int t = threadIdx.x;
    const int lane = t & 31;
    const int wid  = t >> 5;
    const int waveM = wid & 1;
    const int waveN = wid >> 1;
    const int half = lane >> 4;   // 0: K0/K1, 1: K2/K3
    const int lrow = lane & 15;

    const int m_base = blockIdx.x * 128 + waveM * 64;
    const int nblk   = blockIdx.y * 256;           // 256 contiguous (b,l) cols
    const int b      = nblk >> 11;                 // / LL
    const int l0     = nblk & (LL - 1);
    const float* Xb  = X + (size_t)b * CC * LL + l0;   // row k at Xb + k*LL
    const int ncol   = waveN * 64;

    // staging map: 4 issues x 256 threads x 16B cover KT x 256 floats
    const int s_row = t >> 6;          // 0..3
    const int s_col = (t & 63) * 4;    // 0..252

    auto stage = [&](int k0, int buf) {
#pragma unroll
        for (int i = 0; i < 4; ++i) {
            const int r = i * 4 + s_row;
            async_copy_b128(lds_addr_of(&xs[buf][r * 256 + s_col]),
                            Xb + (size_t)(k0 + r) * LL + s_col);
        }
    };

    const v8f zacc = {0.f, 0.f, 0.f, 0.f, 0.f, 0.f, 0.f, 0.f};
    v8f acc[4][4];
#pragma unroll
    for (int i = 0; i < 4; ++i)
#pragma unroll
        for (int j = 0; j < 4; ++j) acc[i][j] = zacc;

    stage(0, 0);

    int buf = 0;
    for (int k0 = 0; k0 < CC; k0 += KT, buf ^= 1) {
        if (k0 + KT < CC) {
            stage(k0 + KT, buf ^ 1);
            // 4 newer ops outstanding; in-order retirement => tile k0 resident
            asm volatile("s_wait_asynccnt 0x4" ::: "memory");
        } else {
            asm volatile("s_wait_asynccnt 0x0" ::: "memory");
        }
        __syncthreads();

#pragma unroll
        for (int kk = 0; kk < KT; kk += 4) {
            const int ka = kk + 2 * half;
            v2f afr[4];
#pragma unroll
            for (int i = 0; i < 4; ++i)
                afr[i] = *(const v2f*)(W + (size_t)(m_base + i * 16 + lrow) * CC
                                         + (k0 + ka));
            v2f bfr[4];
#pragma unroll
            for (int j = 0; j < 4; ++j) {
                const int c = ncol + j * 16 + lrow;
                v2f tt;
                tt.x = xs[buf][ka * 256 + c];
                tt.y = xs[buf][(ka + 1) * 256 + c];
                bfr[j] = tt;
            }
#pragma unroll
            for (int i = 0; i < 4; ++i)
#pragma unroll
                for (int j = 0; j < 4; ++j)
                    acc[i][j] = wmma_f32(afr[i], bfr[j], acc[i][j]);
        }
        __syncthreads();
    }

#pragma unroll
    for (int i = 0; i < 4; ++i) {
#pragma unroll
        for (int j = 0; j < 4; ++j) {
            const int l = l0 + ncol + j * 16 + lrow;
#pragma unroll
            for (int v = 0; v < 8; ++v) {
                const int m = m_base + i * 16 + v + 8 * half;
                Out[(size_t)b * CC * LL + (size_t)m * LL + l] =
                    acc[i][j][v] + bias[m];
            }
        }
    }
}

// ---------------------------------------------------------------------------
// Attention scores: F[b,l,m] = sum_c Theta[b,c,l] * Phi[b,c,m]
// Per batch GEMM: M=LL (l), N=LL (m), K=CC (c). Both operands K-major, so
// fragment loads are lane-coalesced (lanes stride along L). Wave tile 64x64.
// ---------------------------------------------------------------------------
__global__ __launch_bounds__(256) void attn_scores(
    const float* __restrict__ Theta, const float* __restrict__ Phi,
    float* __restrict__ F) {
    const int b = blockIdx.z;
    const int lane = threadIdx.x & 31;
    const int wid  = threadIdx.x >> 5;
    const int waveM = wid & 1;
    const int waveN = wid >> 1;
    const int half = lane >> 4;
    const int lrow = lane & 15;

    const float* Tb = Theta + (size_t)b * CC * LL;
    const float* Pb = Phi   + (size_t)b * CC * LL;
    float*       Fb = F     + (size_t)b * LL * LL;

    const int l_base = blockIdx.x * 128 + waveM * 64;
    const int m_base = blockIdx.y * 256 + waveN * 64;

    const v8f zacc = {0.f, 0.f, 0.f, 0.f, 0.f, 0.f, 0.f, 0.f};
    v8f acc[4][4];
#pragma unroll
    for (int i = 0; i < 4; ++i)
#pragma unroll
        for (int j = 0; j < 4; ++j) acc[i][j] = zacc;

    for (int k = 0; k < CC; k += 4) {
        const int ka = k + 2 * half;
        v2f afr[4];
#pragma unroll
        for (int i = 0; i < 4; ++i) {
            const float* tp = Tb + (size_t)ka * LL + (l_base + i * 16 + lrow);
            v2f tt; tt.x = tp[0]; tt.y = tp[LL];
            afr[i] = tt;
        }
        v2f bfr[4];
#pragma unroll
        for (int j = 0; j < 4; ++j) {
            const float* pp = Pb + (size_t)ka * LL + (m_base + j * 16 + lrow);
            v2f tt; tt.x = pp[0]; tt.y = pp[LL];
            bfr[j] = tt;
        }
#pragma unroll
        for (int i = 0; i < 4; ++i)
#pragma unroll
            for (int j = 0; j < 4; ++j)
                acc[i][j] = wmma_f32(afr[i], bfr[j], acc[i][j]);
    }

#pragma unroll
    for (int i = 0; i < 4; ++i)
#pragma unroll
        for (int j = 0; j < 4; ++j)
#pragma unroll
            for (int v = 0; v < 8; ++v) {
                const int l = l_base + i * 16 + v + 8 * half;
                const int m = m_base + j * 16 + lrow;
                Fb[(size_t)l * LL + m] = acc[i][j][v];
            }
}

// ---------------------------------------------------------------------------
// Row softmax over last axis of F (B*LL rows of length LL), in place.
// ---------------------------------------------------------------------------
__global__ __launch_bounds__(256) void softmax_rows(float* __restrict__ F) {
    float* p = F + (size_t)blockIdx.x * LL;
    const int t = threadIdx.x;
    __shared__ float red[256];

    float vals[8];
    float mx = -INFINITY;
#pragma unroll
    for (int i = 0; i < 8; ++i) {
        vals[i] = p[t + i * 256];
        mx = fmaxf(mx, vals[i]);
    }
    red[t] = mx;
    __syncthreads();
    for (int s = 128; s > 0; s >>= 1) {
        if (t < s) red[t] = fmaxf(red[t], red[t + s]);
        __syncthreads();
    }
    mx = red[0];
    __syncthreads();

    float sum = 0.f;
#pragma unroll
    for (int i = 0; i < 8; ++i) {
        vals[i] = __expf(vals[i] - mx);
        sum += vals[i];
    }
    red[t] = sum;
    __syncthreads();
    for (int s = 128; s > 0; s >>= 1) {
        if (t < s) red[t] += red[t + s];
        __syncthreads();
    }
    const float inv = 1.f / red[0];
#pragma unroll
    for (int i = 0; i < 8; ++i) p[t + i * 256] = vals[i] * inv;
}

// ---------------------------------------------------------------------------
// Apply: Y[b,c,l] = sum_m G[b,c,m] * FS[b,l,m]
// Per batch GEMM: M=CC (c), N=LL (l), K=LL (m).
// FS (the 256 MiB stream) is staged in LDS via double-buffered async copies:
// tile = 256 l-rows x KT2 k-cols, rows padded to FS_PAD floats so ds_load_b64
// fragment reads are bank-conflict free. G stays direct (L2-hot) + prefetch.
// ---------------------------------------------------------------------------
__global__ __launch_bounds__(256) void attn_apply(
    const float* __restrict__ G, const float* __restrict__ FS,
    float* __restrict__ Y) {
    __shared__ float fs[2][256 * FS_PAD];   // 2 x 20 KB

    const int t = threadIdx.x;
    const int lane = t & 31;
    const int wid  = t >> 5;
    const int waveM = wid & 1;
    const int waveN = wid >> 1;
    const int half = lane >> 4;
    const int lrow = lane & 15;
    const int b = blockIdx.z;

    const float* Gb = G  + (size_t)b * CC * LL;
    const float* Fb = FS + (size_t)b * LL * LL;
    float*       Yb = Y  + (size_t)b * CC * LL;

    const int c_base = blockIdx.x * 128 + waveM * 64;
    const int lblk   = blockIdx.y * 256;   // block's 256 l-rows
    const int l_loc  = waveN * 64;         // wave's l offset within tile

    // staging map: 4 issues; each issue: 64 rows x 16 floats (4 thr/row x b128)
    const int s_row = t >> 2;          // 0..63
    const int s_col = (t & 3) * 4;     // 0,4,8,12

    auto stage = [&](int k0, int buf) {
#pragma unroll
        for (int i = 0; i < 4; ++i) {
            const int r = i * 64 + s_row;
            async_copy_b128(lds_addr_of(&fs[buf][r * FS_PAD + s_col]),
                            Fb + (size_t)(lblk + r) * LL + (k0 + s_col));
        }
    };

    const v8f zacc = {0.f, 0.f, 0.f, 0.f, 0.f, 0.f, 0.f, 0.f};
    v8f acc[4][4];
#pragma unroll
    for (int i = 0; i < 4; ++i)
#pragma unroll
        for (int j = 0; j < 4; ++j) acc[i][j] = zacc;

    stage(0, 0);

    int buf = 0;
    for (int k0 = 0; k0 < LL; k0 += KT2, buf ^= 1) {
        if (k0 + KT2 < LL) {
            stage(k0 + KT2, buf ^ 1);
            asm volatile("s_wait_asynccnt 0x4" ::: "memory");
        } else {
            asm volatile("s_wait_asynccnt 0x0" ::: "memory");
        }
        __syncthreads();

        if (k0 + 512 < LL) {
            // keep the G stream warm (global_prefetch_b8)
#pragma unroll
            for (int i = 0; i < 4; ++i)
                __builtin_prefetch(Gb + (size_t)(c_base + i * 16 + lrow) * LL
                                      + (k0 + 512), 0, 1);
        }

#pragma unroll
        for (int kk = 0; kk < KT2; kk += 4) {
            const int ka = kk + 2 * half;
            v2f afr[4];
#pragma unroll
            for (int i = 0; i < 4; ++i)
                afr[i] = *(const v2f*)(Gb + (size_t)(c_base + i * 16 + lrow) * LL
                                          + (k0 + ka));
            v2f bfr[4];
#pragma unroll
            for (int j = 0; j < 4; ++j)
                bfr[j] = *(const v2f*)&fs[buf][(l_loc + j * 16 + lrow) * FS_PAD + ka];
#pragma unroll
            for (int i = 0; i < 4; ++i)
#pragma unroll
                for (int j = 0; j < 4; ++j)
                    acc[i][j] = wmma_f32(afr[i], bfr[j], acc[i][j]);
        }
        __syncthreads();
    }

#pragma unroll
    for (int i = 0; i < 4; ++i)
#pragma unroll
        for (int j = 0; j < 4; ++j)
#pragma unroll
            for (int v = 0; v < 8; ++v) {
                const int c = c_base + i * 16 + v + 8 * half;
                const int l = lblk + l_loc + j * 16 + lrow;
                Yb[(size_t)c * LL + l] = acc[i][j][v];
            }
}

// ---------------------------------------------------------------------------
// BatchNorm stats: per channel mean & rsqrt(var+eps) over (B, L).
// ---------------------------------------------------------------------------
__global__ __launch_bounds__(256) void bn_stats(
    const float* __restrict__ Wy, float* __restrict__ stats) {
    const int c = blockIdx.x;
    const int t = threadIdx.x;
    float s = 0.f, s2 = 0.f;
    for (int b = 0; b < BB; ++b) {
        const float* p = Wy + (size_t)b * CC * LL + (size_t)c * LL;
        for (int l = t; l < LL; l += 256) {
            float v = p[l];
            s += v;
            s2 += v * v;
        }
    }
    __shared__ float r1[256], r2[256];
    r1[t] = s; r2[t] = s2;
    __syncthreads();
    for (int st = 128; st > 0; st >>= 1) {
        if (t < st) { r1[t] += r1[t + st]; r2[t] += r2[t + st]; }
        __syncthreads();
    }
    if (t == 0) {
        const float n = (float)(BB * LL);
        float mean = r1[0] / n;
        float var  = r2[0] / n - mean * mean;
        stats[c]      = mean;
        stats[CC + c] = rsqrtf(var + BN_EPS);
    }
}

// ---------------------------------------------------------------------------
// BN apply + residual: out = gamma*(Wy-mu)*inv + beta + x
// ---------------------------------------------------------------------------
__global__ __launch_bounds__(256) void bn_apply(
    const float* __restrict__ Wy, const float* __restrict__ Xin,
    const float* __restrict__ gamma, const float* __restrict__ beta,
    const float* __restrict__ stats, float* __restrict__ Outp) {
    size_t i = (size_t)blockIdx.x * blockDim.x + threadIdx.x;
    const size_t total = (size_t)BB * CC * LL;
    if (i < total) {
        int c = (int)((i / LL) % CC);
        float mean = stats[c];
        float inv  = stats[CC + c];
        Outp[i] = gamma[c] * (Wy[i] - mean) * inv + beta[c] + Xin[i];
    }
}

extern "C" void kernel_launch(void* const* d_in, const int* in_sizes, int n_in,
                              void* d_out, int out_size, void* d_ws, size_t ws_size,
                              hipStream_t stream) {
    const float* x     = (const float*)d_in[0];
    const float* Wg    = (const float*)d_in[1];
    const float* bg    = (const float*)d_in[2];
    const float* Wt    = (const float*)d_in[3];
    const float* bt    = (const float*)d_in[4];
    const float* Wp    = (const float*)d_in[5];
    const float* bp    = (const float*)d_in[6];
    const float* Ww    = (const float*)d_in[7];
    const float* bw    = (const float*)d_in[8];
    const float* gamma = (const float*)d_in[9];
    const float* beta  = (const float*)d_in[10];
    float* out = (float*)d_out;

    // Workspace partition (floats):
    //   g | theta (-> y) | phi (-> Wy) | f | stats
    float* ws    = (float*)d_ws;
    float* g     = ws;
    float* theta = ws + (size_t)8388608;
    float* phi   = ws + (size_t)16777216;
    float* f     = ws + (size_t)25165824;
    float* stats = ws + (size_t)92274688;
    float* y  = theta;  // reuse after scores consumed theta
    float* Wy = phi;    // reuse after scores consumed phi

    const dim3 blk(256, 1, 1);
    const dim3 grid_pw(CC / 128, (BB * LL) / 256, 1);       // 2 x 128
    const dim3 grid_sc(LL / 128, LL / 256, BB);             // 16 x 8 x 16
    const dim3 grid_ap(CC / 128, LL / 256, BB);             // 2 x 8 x 16

    pointwise_gemm<<<grid_pw, blk, 0, stream>>>(x, Wg, bg, g);
    pointwise_gemm<<<grid_pw, blk, 0, stream>>>(x, Wt, bt, theta);
    pointwise_gemm<<<grid_pw, blk, 0, stream>>>(x, Wp, bp, phi);

    attn_scores<<<grid_sc, blk, 0, stream>>>(theta, phi, f);

    softmax_rows<<<dim3(BB * LL, 1, 1), blk, 0, stream>>>(f);

    attn_apply<<<grid_ap, blk, 0, stream>>>(g, f, y);

    pointwise_gemm<<<grid_pw, blk, 0, stream>>>(y, Ww, bw, Wy);

    bn_stats<<<dim3(CC, 1, 1), blk, 0, stream>>>(Wy, stats);
    bn_apply<<<dim3((BB * CC * LL) / 256, 1, 1), blk, 0, stream>>>(
        Wy, x, gamma, beta, stats, out);
}